// HeteroSAGEFull_19258633355705
// MI455X (gfx1250) — compile-verified
//
#include <hip/hip_runtime.h>
#include <hip/hip_bf16.h>

typedef float v2f __attribute__((ext_vector_type(2)));
typedef float v8f __attribute__((ext_vector_type(8)));

#define BPITCH 144  // pair-row pitch in float2 units (288 dwords == 32 mod 64 banks)

// ---------------- degree count ----------------
__global__ __launch_bounds__(256)
void k_count(const int* __restrict__ ei, float* __restrict__ cnt, int E) {
  int e = blockIdx.x * 256 + threadIdx.x;
  if (e < E) atomicAdd(&cnt[ei[E + e]], 1.0f);
}

__global__ __launch_bounds__(256)
void k_recip(float* __restrict__ f, int n) {
  int i = blockIdx.x * 256 + threadIdx.x;
  if (i < n) f[i] = 1.0f / fmaxf(f[i], 1.0f);
}

// ---------------- sum up to 3 weight arrays, scaled ----------------
__global__ __launch_bounds__(256)
void k_wsum3(float* __restrict__ o, const float* __restrict__ a,
             const float* __restrict__ b, const float* __restrict__ c,
             int n, float scale) {
  int i = blockIdx.x * 256 + threadIdx.x;
  if (i < n) {
    float v = a[i];
    if (b) v += b[i];
    if (c) v += c[i];
    o[i] = v * scale;
  }
}

// ---------------- edge scatter-add (float4 per thread) ----------------
__global__ __launch_bounds__(256)
void k_scatter4(const int* __restrict__ ei, const float* __restrict__ x,
                float* __restrict__ agg, int E, int f4sh) {
  int gid = blockIdx.x * 256 + threadIdx.x;
  int total = E << f4sh;
  if (gid >= total) return;
  int e = gid >> f4sh;
  int j = gid & ((1 << f4sh) - 1);
  int s = ei[e];
  int d = ei[E + e];
  const float4 v = ((const float4*)x)[(((size_t)s) << f4sh) + j];
  float* p = agg + (((((size_t)d) << f4sh) + j) << 2);
  atomicAdd(p + 0, v.x); atomicAdd(p + 1, v.y);
  atomicAdd(p + 2, v.z); atomicAdd(p + 3, v.w);
}

// ---------------- WMMA f32 GEMM: C[M,128] (+)= ascale*diag(rs)*A[M,K] @ W[K,128] (+bias) ----------------
// K is a compile-time constant (32/64/128): full unroll, branchless inner loop.
// B tile staged in LDS in fragment-native layout: pair-row p holds
// {W[2p][col], W[2p+1][col]} as one float2 -> each B fragment is one ds_load_b64
// (pairs of fragments merge into ds_load_2addr_b64). All 8 fragments of a
// k-step are fetched into distinct registers before the 8 WMMAs, so the DS
// waits stagger and later-load latency hides behind earlier WMMAs.
template <int K>
__global__ __launch_bounds__(256)
void k_gemm_n128(const float* __restrict__ A, const float* __restrict__ W,
                 const float* __restrict__ rs, float ascale,
                 const float* __restrict__ bias, float* __restrict__ C,
                 int M, int accumulate, int relu) {
  __shared__ float2 Bsh[(K / 2) * BPITCH];
  // cooperative load of W [K,128] row-major -> pair-interleaved LDS
  {
    float* lds = (float*)Bsh;
    for (int i = threadIdx.x; i < K * 32; i += 256) {
      int row  = i >> 5;
      int colq = (i & 31) << 2;
      const float4 w = ((const float4*)W)[i];
      int base = ((row >> 1) * BPITCH + colq) * 2 + (row & 1);
      lds[base + 0] = w.x;
      lds[base + 2] = w.y;
      lds[base + 4] = w.z;
      lds[base + 6] = w.w;
    }
  }
  __syncthreads();

  const int lane  = threadIdx.x & 31;
  const int wave  = threadIdx.x >> 5;
  const int half  = lane >> 4;   // 0: lanes 0-15, 1: lanes 16-31
  const int l16   = lane & 15;
  const int mBase = blockIdx.x * 128 + wave * 16;
  const int m     = mBase + l16;             // A row this lane feeds
  const bool mvalid = (m < M);
  const int  mc   = mvalid ? m : 0;          // clamped: load stays in-bounds
  // zero scale for invalid rows -> contributes nothing; no branches in the loop
  const float arow = mvalid ? (ascale * (rs ? rs[mc] : 1.0f)) : 0.0f;
  const float2* __restrict__ Arow = (const float2*)(A + (size_t)mc * K);

  v8f acc[8];
#pragma unroll
  for (int nt = 0; nt < 8; ++nt) {
    int colg = nt * 16 + l16;
#pragma unroll
    for (int v = 0; v < 8; ++v) {
      int row = mBase + v + half * 8;
      int rc  = (row < M) ? row : 0;         // clamp; garbage never stored
      float c0 = 0.0f;
      if (accumulate) c0 = C[(size_t)rc * 128 + colg];
      else if (bias)  c0 = bias[colg];
      acc[nt][v] = c0;
    }
  }

#pragma unroll
  for (int k = 0; k < K; k += 4) {
    const float2 t = Arow[(k >> 1) + half]; // A[mc][k+2*half], A[mc][k+2*half+1]
    v2f a;
    a[0] = t.x * arow;
    a[1] = t.y * arow;
    const int prow = (k >> 1) + half;       // pair-row: B rows {k+2h, k+2h+1}
    // fetch all 8 B fragments first (distinct regs -> batched DS, staggered waits)
    float2 bb[8];
#pragma unroll
    for (int nt = 0; nt < 8; ++nt)
      bb[nt] = Bsh[prow * BPITCH + nt * 16 + l16];
#pragma unroll
    for (int nt = 0; nt < 8; ++nt) {
      v2f b;
      b[0] = bb[nt].x;
      b[1] = bb[nt].y;
      acc[nt] = __builtin_amdgcn_wmma_f32_16x16x4_f32(
          false, a, false, b, (short)0, acc[nt], false, false);
    }
  }

#pragma unroll
  for (int nt = 0; nt < 8; ++nt) {
    int colg = nt * 16 + l16;
#pragma unroll
    for (int v = 0; v < 8; ++v) {
      int row = mBase + v + half * 8;
      if (row < M) {
        float val = acc[nt][v];
        if (relu) val = fmaxf(val, 0.0f);
        C[(size_t)row * 128 + colg] = val;
      }
    }
  }
}

// ---------------- final FC: out[M,6] = h[M,128] @ W[128,6] + b ----------------
__global__ __launch_bounds__(256)
void k_fc(const float* __restrict__ h, const float* __restrict__ W,
          const float* __restrict__ b, float* __restrict__ out, int M) {
  __shared__ float Wsh[128 * 6];
  __shared__ float bsh[6];
  for (int i = threadIdx.x; i < 768; i += 256) Wsh[i] = W[i];
  if (threadIdx.x < 6) bsh[threadIdx.x] = b[threadIdx.x];
  __syncthreads();
  int r = blockIdx.x * 256 + threadIdx.x;
  if (r >= M) return;
  float acc[6] = {0, 0, 0, 0, 0, 0};
  const float4* hr = (const float4*)(h + (size_t)r * 128);
#pragma unroll 4
  for (int k4 = 0; k4 < 32; ++k4) {
    float4 hv = hr[k4];
    int k = k4 * 4;
#pragma unroll
    for (int o = 0; o < 6; ++o) acc[o] += hv.x * Wsh[(k + 0) * 6 + o];
#pragma unroll
    for (int o = 0; o < 6; ++o) acc[o] += hv.y * Wsh[(k + 1) * 6 + o];
#pragma unroll
    for (int o = 0; o < 6; ++o) acc[o] += hv.z * Wsh[(k + 2) * 6 + o];
#pragma unroll
    for (int o = 0; o < 6; ++o) acc[o] += hv.w * Wsh[(k + 3) * 6 + o];
  }
#pragma unroll
  for (int o = 0; o < 6; ++o) out[(size_t)r * 6 + o] = acc[o] + bsh[o];
}

// ---------------- GEMM dispatch on K ----------------
static inline void launch_gemm(const float* A, const float* W, const float* rs,
                               float ascale, const float* bias, float* C,
                               int M, int K, int accumulate, int relu,
                               hipStream_t stream) {
  dim3 g((M + 127) / 128), b(256);
  if (K == 128)
    k_gemm_n128<128><<<g, b, 0, stream>>>(A, W, rs, ascale, bias, C, M, accumulate, relu);
  else if (K == 64)
    k_gemm_n128<64><<<g, b, 0, stream>>>(A, W, rs, ascale, bias, C, M, accumulate, relu);
  else
    k_gemm_n128<32><<<g, b, 0, stream>>>(A, W, rs, ascale, bias, C, M, accumulate, relu);
}

// =====================================================================

extern "C" void kernel_launch(void* const* d_in, const int* in_sizes, int n_in,
                              void* d_out, int out_size, void* d_ws, size_t ws_size,
                              hipStream_t stream) {
  (void)ws_size; (void)out_size;
  if (n_in < 79) return;

  // node types: 0=plot 1=building 2=intersection 3=street
  const int FT[4] = {64, 64, 32, 32};
  int NT[4];
  NT[0] = in_sizes[0] / 64; NT[1] = in_sizes[1] / 64;
  NT[2] = in_sizes[2] / 32; NT[3] = in_sizes[3] / 32;
  const int E = in_sizes[4] / 2;

  struct RelT { int s, d, ei; };
  const RelT R[10] = {{0,0,4},{1,1,5},{2,3,6},{3,2,7},{0,1,8},
                      {1,0,9},{0,3,10},{3,0,11},{1,3,12},{3,1,13}};
  const int relCnt[4]      = {3, 3, 1, 3};          // relations per dst type
  const int lastRel[4]     = {7, 9, 3, 8};          // last ETS rel targeting each dst
  const int dstRels[4][3]  = {{0,5,7},{1,4,9},{3,-1,-1},{2,6,8}};

  // ---- locate params (handle jax-pytree sorted vs dict-insertion flattening) ----
  const int pbase = 14;
  const bool leafIns  = (in_sizes[pbase + 1] == 128);   // (Wl,bl,Wr) vs (Wl,Wr,bl)
  const int  wrOff    = leafIns ? 2 : 1;
  const int  blOff    = leafIns ? 1 : 2;
  const bool relAlpha = (in_sizes[pbase + 6] == 8192);  // alphabetical rel order?
  const int  a2e[10]  = {1, 5, 8, 2, 4, 0, 6, 9, 3, 7};

  const float *Wl1[10], *Wr1[10], *bl1[10], *Wl2[10], *Wr2[10], *bl2[10];
  for (int j = 0; j < 10; ++j) {
    int rid = relAlpha ? a2e[j] : j;
    Wl1[rid] = (const float*)d_in[pbase + 3*j];
    Wr1[rid] = (const float*)d_in[pbase + 3*j + wrOff];
    bl1[rid] = (const float*)d_in[pbase + 3*j + blOff];
    Wl2[rid] = (const float*)d_in[pbase + 30 + 3*j];
    Wr2[rid] = (const float*)d_in[pbase + 30 + 3*j + wrOff];
    bl2[rid] = (const float*)d_in[pbase + 30 + 3*j + blOff];
  }
  const float* Wl3 = (const float*)d_in[pbase + 60];
  const float* Wr3 = (const float*)d_in[pbase + 60 + wrOff];
  const float* bl3 = (const float*)d_in[pbase + 60 + blOff];
  const float* Wfc = (const float*)d_in[pbase + 63];
  const float* bfc = (const float*)d_in[pbase + 64];

  const float* X[4] = {(const float*)d_in[0], (const float*)d_in[1],
                       (const float*)d_in[2], (const float*)d_in[3]};
  const int* EI[10];
  for (int r = 0; r < 10; ++r) EI[r] = (const int*)d_in[R[r].ei];

  // ---- carve workspace ----
  char* wsb = (char*)d_ws;
  size_t off = 0;
  auto alloc = [&](size_t bytes) -> float* {
    size_t a = (off + 255) & ~(size_t)255;
    off = a + bytes;
    return (float*)(wsb + a);
  };
  float *h1[4], *h2[4];
  for (int t = 0; t < 4; ++t) h1[t] = alloc((size_t)NT[t] * 128 * 4);
  for (int t = 0; t < 4; ++t) h2[t] = alloc((size_t)NT[t] * 128 * 4);
  float* agg = alloc((size_t)NT[1] * 128 * 4);  // biggest aggregation tile, reused
  size_t cntTot = 0; for (int r = 0; r < 10; ++r) cntTot += NT[R[r].d];
  float* cntBlk = alloc(cntTot * 4);
  float* rinv[10];
  { size_t o2 = 0; for (int r = 0; r < 10; ++r) { rinv[r] = cntBlk + o2; o2 += NT[R[r].d]; } }
  float *wrS1[4], *blS1[4], *wrS2[4], *blS2[4];
  for (int t = 0; t < 4; ++t) {
    wrS1[t] = alloc((size_t)FT[t] * 128 * 4); blS1[t] = alloc(128 * 4);
    wrS2[t] = alloc(128 * 128 * 4);           blS2[t] = alloc(128 * 4);
  }

  // ---- degrees (shared by all layers) -> rinv = 1/max(cnt,1) ----
  hipMemsetAsync(cntBlk, 0, cntTot * 4, stream);
  for (int r = 0; r < 10; ++r)
    k_count<<<dim3((E + 255) / 256), dim3(256), 0, stream>>>(EI[r], rinv[r], E);
  k_recip<<<dim3((int)((cntTot + 255) / 256)), dim3(256), 0, stream>>>(cntBlk, (int)cntTot);

  // ---- pre-sum Wr / bl per dst type, scaled by 1/R_d ----
  for (int t = 0; t < 4; ++t) {
    const float *w1[3] = {nullptr, nullptr, nullptr}, *b1[3] = {nullptr, nullptr, nullptr};
    const float *w2[3] = {nullptr, nullptr, nullptr}, *b2[3] = {nullptr, nullptr, nullptr};
    int nr = relCnt[t];
    float scale = 1.0f / (float)nr;
    for (int q = 0; q < nr; ++q) {
      int r = dstRels[t][q];
      w1[q] = Wr1[r]; b1[q] = bl1[r];
      w2[q] = Wr2[r]; b2[q] = bl2[r];
    }
    int n1 = FT[t] * 128;
    k_wsum3<<<dim3((n1 + 255) / 256), dim3(256), 0, stream>>>(wrS1[t], w1[0], w1[1], w1[2], n1, scale);
    k_wsum3<<<dim3(1), dim3(256), 0, stream>>>(blS1[t], b1[0], b1[1], b1[2], 128, scale);
    k_wsum3<<<dim3((16384 + 255) / 256), dim3(256), 0, stream>>>(wrS2[t], w2[0], w2[1], w2[2], 16384, scale);
    k_wsum3<<<dim3(1), dim3(256), 0, stream>>>(blS2[t], b2[0], b2[1], b2[2], 128, scale);
  }

  // ---- one hetero-SAGE layer ----
  auto runLayer = [&](const float* const* xin, const int* Fin, float* const* xout,
                      const float* const* Wl, float* const* wrS, float* const* blS) {
    for (int t = 0; t < 4; ++t) {
      launch_gemm(xin[t], wrS[t], nullptr, 1.0f, blS[t], xout[t],
                  NT[t], Fin[t], 0, 0, stream);
    }
    for (int r = 0; r < 10; ++r) {
      int s = R[r].s, d = R[r].d;
      int Fs = Fin[s];
      hipMemsetAsync(agg, 0, (size_t)NT[d] * Fs * 4, stream);
      int f4sh = (Fs == 128) ? 5 : (Fs == 64) ? 4 : 3;
      int total = E << f4sh;
      k_scatter4<<<dim3((total + 255) / 256), dim3(256), 0, stream>>>(EI[r], xin[s], agg, E, f4sh);
      launch_gemm(agg, Wl[r], rinv[r], 1.0f / (float)relCnt[d], nullptr, xout[d],
                  NT[d], Fs, 1, (r == lastRel[d]) ? 1 : 0, stream);
    }
  };

  const int F1[4] = {64, 64, 32, 32};
  const int F2[4] = {128, 128, 128, 128};
  runLayer(X, F1, h1, Wl1, wrS1, blS1);
  const float* h1c[4] = {h1[0], h1[1], h1[2], h1[3]};
  runLayer(h1c, F2, h2, Wl2, wrS2, blS2);

  // ---- layer 3: building <- building only ----
  float* h3 = h1[1];  // reuse building h1 region
  launch_gemm(h2[1], Wr3, nullptr, 1.0f, bl3, h3, NT[1], 128, 0, 0, stream);
  hipMemsetAsync(agg, 0, (size_t)NT[1] * 128 * 4, stream);
  {
    int total = E << 5;
    k_scatter4<<<dim3((total + 255) / 256), dim3(256), 0, stream>>>(EI[1], h2[1], agg, E, 5);
  }
  launch_gemm(agg, Wl3, rinv[1], 1.0f, nullptr, h3, NT[1], 128, 1, 1, stream);

  // ---- final FC ----
  k_fc<<<dim3((NT[1] + 255) / 256), dim3(256), 0, stream>>>(h3, Wfc, bfc, (float*)d_out, NT[1]);
}